// RUNG_combined_26706106646708
// MI455X (gfx1250) — compile-verified
//
#include <hip/hip_runtime.h>

// ---------------------------------------------------------------------------
// RUNG propagation, sparsity-exploiting implementation for MI455X (gfx1250).
//  - dense A read only in preprocessing (degrees + CSR edge extraction)
//  - MLP GEMMs via V_WMMA_F32_16X16X4_F32, register-blocked 16x64 per wave,
//    B loads batched per K-step so WMMAs issue back-to-back
//  - per-step: row unit-norm, edge-parallel cosine distance, exact masked
//    quantile via 4-pass radix select on float bits, SCAD-weighted SpMM
// ---------------------------------------------------------------------------

#define N_NODES   8192
#define D_IN_     512
#define HID_      256
#define C_OUT_    64
#define CAP_E     (1 << 19)     // edge capacity (expected ~270K)
#define PROP_STEP 10
#define SCAD_A_F  3.7f
#define EPS_F     1e-8f
#define LAM_F     ((float)(1.0 / 0.9 - 1.0))

typedef __attribute__((ext_vector_type(2))) float v2f;
typedef __attribute__((ext_vector_type(8))) float v8f;

static __device__ __forceinline__ float scad_w(float y, float lam) {
  float al = SCAD_A_F * lam;
  float rp = (y <= lam) ? lam : ((y <= al) ? (al - y) / (SCAD_A_F - 1.0f) : 0.0f);
  return rp / (2.0f * y);
}

// ---------------- preprocessing -------------------------------------------

// one block per row: degree (A has 0/1 entries -> exact integer sums in f32)
__global__ void k_degrees(const float* __restrict__ A, float* Dg, float* Dsq,
                          float* dinv, int* rowcnt) {
  int i = blockIdx.x;
  __shared__ float sd[256];
  float s = 0.0f;
  for (int j = threadIdx.x; j < N_NODES; j += 256) s += A[(size_t)i * N_NODES + j];
  sd[threadIdx.x] = s;
  __syncthreads();
  for (int o = 128; o > 0; o >>= 1) {
    if (threadIdx.x < o) sd[threadIdx.x] += sd[threadIdx.x + o];
    __syncthreads();
  }
  if (threadIdx.x == 0) {
    float deg = sd[0] + 1.0f;              // self loop
    Dg[i] = deg;
    float ds = sqrtf(deg);
    Dsq[i] = ds;
    dinv[i] = 1.0f / ds;
    rowcnt[i] = (int)(sd[0] + 0.5f);       // off-diag nonzeros (entries are 1.0)
  }
}

// exclusive scan of 8192 row counts -> row_ptr[8193], nE
__global__ void k_scan(const int* __restrict__ rowcnt, int* row_ptr, int* nE) {
  __shared__ int part[256];
  int t = threadIdx.x;
  int base = t * 32;
  int s = 0;
  for (int k = 0; k < 32; ++k) s += rowcnt[base + k];
  part[t] = s;
  __syncthreads();
  if (t == 0) {
    int run = 0;
    for (int i = 0; i < 256; ++i) { int v = part[i]; part[i] = run; run += v; }
    row_ptr[N_NODES] = run;
    *nE = run;
  }
  __syncthreads();
  int off = part[t];
  for (int k = 0; k < 32; ++k) { row_ptr[base + k] = off; off += rowcnt[base + k]; }
}

// one wave per row: ordered (deterministic) CSR fill via ballot prefix
__global__ void k_fill(const float* __restrict__ A, const int* __restrict__ row_ptr,
                       const float* __restrict__ dinv, int* ecol, int* erow, float* ev) {
  int row = blockIdx.x * 8 + (threadIdx.x >> 5);
  if (row >= N_NODES) return;
  int lane = threadIdx.x & 31;
  int base = row_ptr[row];
  float di = dinv[row];
  int cur = 0;
  for (int j0 = 0; j0 < N_NODES; j0 += 32) {
    int j = j0 + lane;
    float v = A[(size_t)row * N_NODES + j];
    bool p = (v > 0.0f) && (j != row);
    unsigned long long m = __ballot(p);
    int pre = __popcll(m & ((1ull << lane) - 1ull));
    if (p) {
      int e = base + cur + pre;
      ecol[e] = j;
      erow[e] = row;
      ev[e] = di * dinv[j];                // A_tilde value at edge
    }
    cur += __popcll(m);
  }
}

// ---------------- MLP via f32 WMMA ----------------------------------------

// C[M,N] = act(Amat[M,K] @ Bmat[K,N] + bias)
// one wave computes a 16x64 output strip (4 accumulators). Per K-step: the A
// fragment is loaded once, all 4 B fragments are fetched in one load clause,
// then the 4 WMMAs issue back-to-back (waits cover the whole group).
__global__ void k_gemm_wmma_f32(const float* __restrict__ Am, const float* __restrict__ Bm,
                                const float* __restrict__ bias, float* __restrict__ Cm,
                                int M, int N, int K, int relu) {
  int wave = threadIdx.x >> 5;
  int lane = threadIdx.x & 31;
  int m0 = (blockIdx.x * 4 + wave) * 16;
  int nb = blockIdx.y * 64;                 // 4 consecutive 16-wide N tiles
  if (m0 >= M || nb >= N) return;           // whole-wave uniform exit: EXEC full
  int r = lane & 15;                        // A: M index; B/C: N index within tile
  int h = lane >> 4;                        // lane half selects K pair / M half
  v8f z = {0.f, 0.f, 0.f, 0.f, 0.f, 0.f, 0.f, 0.f};
  v8f acc[4];
  acc[0] = z; acc[1] = z; acc[2] = z; acc[3] = z;
  const float* arow = Am + (size_t)(m0 + r) * K;
  for (int k = 0; k < K; k += 4) {
    __builtin_prefetch(arow + k + 32, 0, 0);       // global_prefetch ahead on A row
    v2f a;
    a[0] = arow[k + 2 * h];                 // A VGPR0: K = 0 (lanes 0-15) / 2 (16-31)
    a[1] = arow[k + 2 * h + 1];             // A VGPR1: K = 1 / 3
    const float* b0p = Bm + (size_t)(k + 2 * h) * N + nb + r;
    const float* b1p = b0p + N;
    v2f b[4];
#pragma unroll
    for (int t = 0; t < 4; ++t) {           // load phase: one clause of 8 b32 loads
      b[t][0] = b0p[16 * t];
      b[t][1] = b1p[16 * t];
    }
#pragma unroll
    for (int t = 0; t < 4; ++t) {           // compute phase: 4 WMMAs back-to-back
      acc[t] = __builtin_amdgcn_wmma_f32_16x16x4_f32(false, a, false, b[t], (short)0,
                                                     acc[t], false, false);
    }
  }
#pragma unroll
  for (int t = 0; t < 4; ++t) {
    int col = nb + 16 * t + r;
    float bv = bias[col];
#pragma unroll
    for (int v = 0; v < 8; ++v) {
      float val = acc[t][v] + bv;
      if (relu) val = fmaxf(val, 0.0f);
      Cm[(size_t)(m0 + v + 8 * h) * N + col] = val;
    }
  }
}

// ---------------- per-iteration kernels -----------------------------------

// U[i,:] = (F[i,:]/Dsq[i]) / max(||F[i,:]/Dsq[i]||, eps)
__global__ void k_unit(const float* __restrict__ F, const float* __restrict__ Dsq,
                       float* __restrict__ U) {
  int i = blockIdx.x;
  int c = threadIdx.x;                      // 64 threads
  __shared__ float sd[64];
  float v = F[(size_t)i * C_OUT_ + c] / Dsq[i];
  sd[c] = v * v;
  __syncthreads();
  for (int o = 32; o > 0; o >>= 1) {
    if (c < o) sd[c] += sd[c + o];
    __syncthreads();
  }
  float nrm = sqrtf(sd[0]);
  U[(size_t)i * C_OUT_ + c] = v / fmaxf(nrm, EPS_F);
}

// one wave per edge: y_e = clip(1 - <U_i, U_j>, 0, 2)
__global__ void k_edge_y(const int* __restrict__ nE, const int* __restrict__ erow,
                         const int* __restrict__ ecol, const float* __restrict__ U,
                         float* __restrict__ ye) {
  int e = blockIdx.x * 8 + (threadIdx.x >> 5);
  if (e >= *nE) return;
  int lane = threadIdx.x & 31;
  int i = erow[e], j = ecol[e];
  float d = U[(size_t)i * C_OUT_ + lane] * U[(size_t)j * C_OUT_ + lane] +
            U[(size_t)i * C_OUT_ + lane + 32] * U[(size_t)j * C_OUT_ + lane + 32];
  for (int o = 16; o > 0; o >>= 1) d += __shfl_xor(d, o);
  if (lane == 0) {
    float y = 1.0f - d;
    ye[e] = fminf(fmaxf(y, 0.0f), 2.0f);
  }
}

// radix-select state setup (+ hist zero). s==0 also derives klo/khi/frac.
__global__ void k_sel_start(const int* __restrict__ nE, unsigned* prefix, int* kcur,
                            int* klo, int* khi, float* frac, unsigned* hist, int s) {
  if (threadIdx.x == 0) {
    if (s == 0) {
      int n = *nE;
      float pos = 0.75f * (float)(n - 1);   // f32, matches reference
      int lo = (int)floorf(pos);
      int hi = (int)ceilf(pos);
      *klo = lo; *khi = hi;
      *frac = pos - (float)lo;
    }
    *prefix = 0u;
    *kcur = (s == 0) ? *klo : *khi;
  }
  hist[threadIdx.x] = 0u;                   // 256 threads
}

// histogram of current 8-bit digit among keys matching the prefix
__global__ void k_hist(const int* __restrict__ nE, const float* __restrict__ ye,
                       const unsigned* __restrict__ prefix, unsigned* hist, int pass) {
  int e = blockIdx.x * 256 + threadIdx.x;
  if (e >= *nE) return;
  unsigned key = __float_as_uint(ye[e]);    // y >= 0 -> bit pattern is order-preserving
  int sh = pass * 8;
  unsigned pf = *prefix;
  bool match = (pass == 3) || ((key >> (sh + 8)) == (pf >> (sh + 8)));
  if (match) atomicAdd(&hist[(key >> sh) & 0xFFu], 1u);
}

// pick digit containing rank k; fold into prefix; re-zero hist
__global__ void k_pick(unsigned* hist, unsigned* prefix, int* kcur, float* vsel,
                       int pass, int s) {
  if (threadIdx.x == 0) {
    int k = *kcur;
    unsigned cum = 0;
    int d = 0;
    for (; d < 256; ++d) {
      unsigned c = hist[d];
      if (cum + c > (unsigned)k) break;
      cum += c;
    }
    if (d > 255) d = 255;
    *prefix |= ((unsigned)d) << (pass * 8);
    *kcur = k - (int)cum;
    if (pass == 0) vsel[s] = __uint_as_float(*prefix);
  }
  __syncthreads();
  hist[threadIdx.x] = 0u;
}

__global__ void k_gamma(const float* __restrict__ vsel, const float* __restrict__ frac,
                        float* gammap, float* lamk) {
  float f = *frac;
  float g = vsel[0] * (1.0f - f) + vsel[1] * f;
  g = fmaxf(g, EPS_F);
  *gammap = g;
  *lamk = g / SCAD_A_F;
}

// precompute SCAD weights once per step (coalesced, edge-parallel) so the
// irregular per-row gather loops below are pure load+FMA
__global__ void k_edge_w(const int* __restrict__ nE, const float* __restrict__ ye,
                         const float* __restrict__ lamk, float* __restrict__ wgt) {
  int e = blockIdx.x * 256 + threadIdx.x;
  if (e >= *nE) return;
  __builtin_prefetch(&ye[e + 2048], 0, 0);
  wgt[e] = scad_w(ye[e], *lamk);
}

// Q_hat[i] = (1 + sum_e w_e) / Dg[i] + LAM   (diag weight = 1 per nan->1 rule)
__global__ void k_qhat(const int* __restrict__ rp, const float* __restrict__ wgt,
                       const float* __restrict__ Dg, float* __restrict__ Qh) {
  int i = blockIdx.x * 256 + threadIdx.x;
  float s = 1.0f;
  int e1 = rp[i + 1];
  for (int e = rp[i]; e < e1; ++e) s += wgt[e];
  Qh[i] = s / Dg[i] + LAM_F;
}

// F_out = (W .* A_tilde) @ F_in / Q + LAM * F0 / Q ; 4 rows per block, col per thread
__global__ void k_spmm(const int* __restrict__ rp, const int* __restrict__ ecol,
                       const float* __restrict__ ev, const float* __restrict__ wgt,
                       const float* __restrict__ F_in, const float* __restrict__ F0,
                       const float* __restrict__ dinv, const float* __restrict__ Qh,
                       float* __restrict__ F_out) {
  int row = blockIdx.x * 4 + (threadIdx.x >> 6);
  int c = threadIdx.x & 63;
  float di = dinv[row];
  float acc = di * di * F_in[(size_t)row * C_OUT_ + c];   // self loop: w=1, Atil=dinv^2
  int e1 = rp[row + 1];
  for (int e = rp[row]; e < e1; ++e) {
    acc += wgt[e] * ev[e] * F_in[(size_t)ecol[e] * C_OUT_ + c];
  }
  float q = Qh[row];
  F_out[(size_t)row * C_OUT_ + c] = acc / q + LAM_F * F0[(size_t)row * C_OUT_ + c] / q;
}

// ---------------- host-side orchestration ---------------------------------

extern "C" void kernel_launch(void* const* d_in, const int* in_sizes, int n_in,
                              void* d_out, int out_size, void* d_ws, size_t ws_size,
                              hipStream_t stream) {
  (void)in_sizes; (void)n_in; (void)out_size; (void)ws_size;
  const float* A  = (const float*)d_in[0];
  const float* X  = (const float*)d_in[1];
  const float* W1 = (const float*)d_in[2];
  const float* b1 = (const float*)d_in[3];
  const float* W2 = (const float*)d_in[4];
  const float* b2 = (const float*)d_in[5];
  float* out = (float*)d_out;

  char* w = (char*)d_ws;
  size_t off = 0;
  auto alloc = [&](size_t bytes) -> void* {
    void* p = w + off;
    off = (off + bytes + 255) & ~(size_t)255;
    return p;
  };
  float*    Dg      = (float*)alloc(N_NODES * sizeof(float));
  float*    Dsq     = (float*)alloc(N_NODES * sizeof(float));
  float*    dinv    = (float*)alloc(N_NODES * sizeof(float));
  float*    Qh      = (float*)alloc(N_NODES * sizeof(float));
  int*      rowcnt  = (int*)alloc(N_NODES * sizeof(int));
  int*      row_ptr = (int*)alloc((N_NODES + 1) * sizeof(int));
  int*      nE      = (int*)alloc(sizeof(int));
  unsigned* hist    = (unsigned*)alloc(256 * sizeof(unsigned));
  unsigned* prefix  = (unsigned*)alloc(sizeof(unsigned));
  int*      kcur    = (int*)alloc(sizeof(int));
  int*      klo     = (int*)alloc(sizeof(int));
  int*      khi     = (int*)alloc(sizeof(int));
  float*    frac    = (float*)alloc(sizeof(float));
  float*    vsel    = (float*)alloc(2 * sizeof(float));
  float*    gammap  = (float*)alloc(sizeof(float));
  float*    lamk    = (float*)alloc(sizeof(float));
  int*      ecol    = (int*)alloc((size_t)CAP_E * sizeof(int));
  int*      erow    = (int*)alloc((size_t)CAP_E * sizeof(int));
  float*    ev      = (float*)alloc((size_t)CAP_E * sizeof(float));
  float*    ye      = (float*)alloc((size_t)CAP_E * sizeof(float));
  float*    wgt     = (float*)alloc((size_t)CAP_E * sizeof(float));
  float*    U       = (float*)alloc((size_t)N_NODES * C_OUT_ * sizeof(float));
  float*    H       = (float*)alloc((size_t)N_NODES * HID_ * sizeof(float));
  float*    F0      = (float*)alloc((size_t)N_NODES * C_OUT_ * sizeof(float));
  float*    fb0     = (float*)alloc((size_t)N_NODES * C_OUT_ * sizeof(float));
  float*    fb1     = (float*)alloc((size_t)N_NODES * C_OUT_ * sizeof(float));

  // ---- preprocessing (only passes over dense A) ----
  k_degrees<<<N_NODES, 256, 0, stream>>>(A, Dg, Dsq, dinv, rowcnt);
  k_scan<<<1, 256, 0, stream>>>(rowcnt, row_ptr, nE);
  k_fill<<<N_NODES / 8, 256, 0, stream>>>(A, row_ptr, dinv, ecol, erow, ev);

  // ---- MLP: H = relu(X@W1+b1); F0 = H@W2+b2 (f32 WMMA, 16x64 per wave) ----
  k_gemm_wmma_f32<<<dim3(N_NODES / 64, HID_ / 64), 128, 0, stream>>>(
      X, W1, b1, H, N_NODES, HID_, D_IN_, 1);
  k_gemm_wmma_f32<<<dim3(N_NODES / 64, 1), 128, 0, stream>>>(
      H, W2, b2, F0, N_NODES, C_OUT_, HID_, 0);

  // ---- propagation ----
  for (int t = 0; t < PROP_STEP; ++t) {
    const float* fin = (t == 0) ? F0 : ((t % 2 == 1) ? fb1 : fb0);
    float* fout = (t == PROP_STEP - 1) ? out : ((t % 2 == 0) ? fb1 : fb0);

    k_unit<<<N_NODES, 64, 0, stream>>>(fin, Dsq, U);
    k_edge_y<<<CAP_E / 8, 256, 0, stream>>>(nE, erow, ecol, U, ye);

    for (int s = 0; s < 2; ++s) {           // lo and hi order statistics
      k_sel_start<<<1, 256, 0, stream>>>(nE, prefix, kcur, klo, khi, frac, hist, s);
      for (int pass = 3; pass >= 0; --pass) {
        k_hist<<<CAP_E / 256, 256, 0, stream>>>(nE, ye, prefix, hist, pass);
        k_pick<<<1, 256, 0, stream>>>(hist, prefix, kcur, vsel, pass, s);
      }
    }
    k_gamma<<<1, 1, 0, stream>>>(vsel, frac, gammap, lamk);
    k_edge_w<<<CAP_E / 256, 256, 0, stream>>>(nE, ye, lamk, wgt);

    k_qhat<<<N_NODES / 256, 256, 0, stream>>>(row_ptr, wgt, Dg, Qh);
    k_spmm<<<N_NODES / 4, 256, 0, stream>>>(row_ptr, ecol, ev, wgt, fin, F0, dinv,
                                            Qh, fout);
  }
}